// Interaction_2018634629402
// MI455X (gfx1250) — compile-verified
//
#include <hip/hip_runtime.h>
#include <math.h>

typedef __attribute__((ext_vector_type(2))) float v2f;
typedef __attribute__((ext_vector_type(4))) float v4f;
typedef __attribute__((ext_vector_type(8))) float v8f;
typedef unsigned int u32x4 __attribute__((ext_vector_type(4)));
typedef int i32x8 __attribute__((ext_vector_type(8)));
typedef int i32x4 __attribute__((ext_vector_type(4)));

#define ALPHA 0.2f
#define BB 16
#define NN 1024
#define MM 2048
#define DD 128
#define KC 64              // k-chunk staged by TDM
#define LROW 68            // LDS row stride in dwords (64 data + 4 TDM pad)

__device__ __forceinline__ float leaky(float x) { return x > 0.f ? x : ALPHA * x; }

// Issue one TDM 2D tile load: [tile1 rows][KC cols], row stride = stride0 elems,
// LDS rows padded to LROW dwords via TDM pad (interval=64 dwords, amount=4 dwords).
__device__ __forceinline__ void tdm_load_2d(unsigned int lds_off, const float* gptr,
                                            int tdim0, int tdim1, int stride0) {
  unsigned long long ga = (unsigned long long)(uintptr_t)gptr;
  u32x4 g0;
  g0[0] = 1u;                                        // count=1 (valid user D#)
  g0[1] = lds_off;                                   // LDS byte address
  g0[2] = (unsigned int)(ga & 0xffffffffu);          // global_addr[31:0]
  g0[3] = (unsigned int)((ga >> 32) & 0x1ffffffu) | (2u << 30);  // addr[56:32] | type=2
  i32x8 g1;
  g1[0] = (2 << 16) | (1 << 20) | (5 << 22) | (3 << 25); // 4B elems, pad_en, 64dw interval, 4dw pad
  g1[1] = (int)((tdim0 & 0xffff) << 16);                 // tensor_dim0[15:0]
  g1[2] = (int)(((unsigned)tdim0 >> 16) | ((tdim1 & 0xffff) << 16));
  g1[3] = (int)(((unsigned)tdim1 >> 16) | (KC << 16));   // tile_dim0 = KC
  g1[4] = (int)(DD & 0xffff);                            // tile_dim1 = 128, tile_dim2 = 0
  g1[5] = stride0;                                       // tensor_dim0_stride[31:0]
  g1[6] = 0;
  g1[7] = 0;
  i32x4 z4 = {0, 0, 0, 0};
  i32x8 z8 = {0, 0, 0, 0, 0, 0, 0, 0};
  __builtin_amdgcn_tensor_load_to_lds(g0, g1, z4, z4, z8, 0);
}

// ---------------- wb = W @ b_vec (tiny) ----------------
__global__ void wb_kernel(const float* __restrict__ Wm, const float* __restrict__ Wp,
                          const float* __restrict__ bvec, float* __restrict__ wbm,
                          float* __restrict__ wbp) {
  int t = threadIdx.x;
  if (t < 128) {
    float a = 0.f;
    for (int e = 0; e < 128; ++e) a += Wm[t * 128 + e] * bvec[e];
    wbm[t] = a;
  } else {
    int d = t - 128;
    float a = 0.f;
    for (int e = 0; e < 128; ++e) a += Wp[d * 128 + e] * bvec[128 + e];
    wbp[d] = a;
  }
}

// ---------------- s[row] = X[row,:] . wb  (one wave per row) ----------------
__global__ __launch_bounds__(256) void s_kernel(const float* __restrict__ X,
                                                const float* __restrict__ wb,
                                                float* __restrict__ s) {
  int wave = threadIdx.x >> 5, lane = threadIdx.x & 31;
  int row = blockIdx.x * 8 + wave;
  const float* x = X + (size_t)row * 128;
  float a = 0.f;
  for (int i = lane; i < 128; i += 32) a += x[i] * wb[i];
  for (int off = 16; off; off >>= 1) a += __shfl_xor(a, off, 32);
  if (lane == 0) s[row] = a;
}

// ---------------- per-batch max of s_p ----------------
__global__ __launch_bounds__(256) void maxp_kernel(const float* __restrict__ sp,
                                                   float* __restrict__ mxp) {
  __shared__ float red[256];
  int b = blockIdx.x;
  float m = -3.0e38f;
  for (int i = threadIdx.x; i < MM; i += 256) m = fmaxf(m, sp[b * MM + i]);
  red[threadIdx.x] = m;
  __syncthreads();
  for (int s = 128; s > 0; s >>= 1) {
    if (threadIdx.x < s) red[threadIdx.x] = fmaxf(red[threadIdx.x], red[threadIdx.x + s]);
    __syncthreads();
  }
  if (threadIdx.x == 0) mxp[b] = red[0];
}

// ---------------- invZ[b,n] = 1 / sum_m exp(leaky(s_m+s_p) - rowmax) ----------------
__global__ __launch_bounds__(256) void z_kernel(const float* __restrict__ sm,
                                                const float* __restrict__ sp,
                                                const float* __restrict__ mxp,
                                                float* __restrict__ izv) {
  __shared__ float sps[MM];
  int b = blockIdx.x >> 2;
  int n = ((blockIdx.x & 3) << 8) + threadIdx.x;
  for (int i = threadIdx.x; i < MM; i += 256) sps[i] = sp[b * MM + i];
  __syncthreads();
  float s = sm[b * NN + n];
  float rm = leaky(s + mxp[b]);
  float z = 0.f;
  for (int m = 0; m < MM; ++m) {
    float x = leaky(s + sps[m]);
    z += __expf(x - rm);
  }
  izv[b * NN + n] = 1.f / z;
}

// ---------------- hT[b][d][row] = sum_k X[b][row][k] * W[k][d] ----------------
__global__ __launch_bounds__(256) void proj_kernel(const float* __restrict__ X,
                                                   const float* __restrict__ W,
                                                   float* __restrict__ hT, int R) {
  __shared__ float wl[128 * 128];  // 64 KB, {W[2q][c],W[2q+1][c]} pairs
  for (int i = threadIdx.x; i < 128 * 128; i += 256) {
    int k = i >> 7, c = i & 127;
    wl[(((k >> 1) * 128 + c) << 1) + (k & 1)] = W[i];
  }
  __syncthreads();
  int wave = threadIdx.x >> 5, lane = threadIdx.x & 31;
  int hi = lane >> 4, lj = lane & 15;
  int tile = blockIdx.x * 8 + wave;
  int rowg = tile * 16;
  int b = rowg / R;
  int n0 = rowg - b * R;
  const float* xr = X + ((size_t)b * R + n0 + lj) * 128 + 2 * hi;
  float* hb = hT + (size_t)b * 128 * R;
  v8f acc[8] = {};
  for (int k = 0; k < 128; k += 4) {
    v2f a = *(const v2f*)(xr + k);
    const float* wp = &wl[(((k >> 1) + hi) * 128) << 1];
#pragma unroll
    for (int t = 0; t < 8; ++t) {
      v2f bf = *(const v2f*)(wp + ((t * 16 + lj) << 1));
      acc[t] = __builtin_amdgcn_wmma_f32_16x16x4_f32(false, a, false, bf, (short)0, acc[t],
                                                     false, false);
    }
  }
#pragma unroll
  for (int t = 0; t < 8; ++t) {
    float* dst = hb + (size_t)(t * 16 + lj) * R + n0 + hi * 8;
    v4f lo = {acc[t][0], acc[t][1], acc[t][2], acc[t][3]};
    v4f hh = {acc[t][4], acc[t][5], acc[t][6], acc[t][7]};
    *(v4f*)dst = lo;
    *(v4f*)(dst + 4) = hh;
  }
}

// ---------------- out_m = elu( softmax-rows @ h_p ) ----------------
// Block: 4 n-tiles x 128 cols; wave w -> tile (w>>1), col-half (w&1).
// B panel (h_pT chunk [128 d][KC m]) staged per block by TDM, double-buffered.
__global__ __launch_bounds__(256) void attn_m_kernel(
    const float* __restrict__ sm, const float* __restrict__ sp,
    const float* __restrict__ mxp, const float* __restrict__ izv,
    const float* __restrict__ hpT, float* __restrict__ outm) {
  __shared__ __align__(16) float bufA[DD * LROW];
  __shared__ __align__(16) float bufB[DD * LROW];
  int wave = threadIdx.x >> 5, lane = threadIdx.x & 31;
  int hi = lane >> 4, lj = lane & 15;
  int b = blockIdx.x >> 4;                       // 16 blocks per batch
  int tloc = (blockIdx.x & 15) * 4 + (wave >> 1);
  int cb0 = (wave & 1) * 4;
  int n0 = tloc * 16;
  const float* spb = sp + b * MM;
  const float* hp = hpT + (size_t)b * 128 * MM;
  float smr = sm[b * NN + n0 + lj];
  float rmax = leaky(smr + mxp[b]);
  unsigned int l0 = (unsigned int)(uintptr_t)bufA;
  unsigned int l1 = (unsigned int)(uintptr_t)bufB;
  if (wave == 0) tdm_load_2d(l0, hp, MM, DD, MM);
  v8f acc[4] = {};
  const int NCH = MM / KC;  // 32
  for (int c = 0; c < NCH; ++c) {
    if (wave == 0) {
      if (c + 1 < NCH) {
        tdm_load_2d(((c + 1) & 1) ? l1 : l0, hp + (c + 1) * KC, MM, DD, MM);
        __builtin_amdgcn_s_wait_tensorcnt(1);
      } else {
        __builtin_amdgcn_s_wait_tensorcnt(0);
      }
    }
    __syncthreads();
    const float* buf = (c & 1) ? bufB : bufA;
    int kbase = c * KC;
    for (int kk = 0; kk < KC; kk += 4) {
      v2f spv = *(const v2f*)(spb + kbase + kk + 2 * hi);
      float x0 = leaky(smr + spv.x);
      float x1 = leaky(smr + spv.y);
      v2f a;
      a.x = __expf(x0 - rmax);
      a.y = __expf(x1 - rmax);
#pragma unroll
      for (int t = 0; t < 4; ++t) {
        v2f bf = *(const v2f*)(buf + ((cb0 + t) * 16 + lj) * LROW + kk + 2 * hi);
        acc[t] = __builtin_amdgcn_wmma_f32_16x16x4_f32(false, a, false, bf, (short)0,
                                                       acc[t], false, false);
      }
    }
    __syncthreads();
  }
  const float* zp = izv + b * NN + n0 + hi * 8;
  float z[8];
#pragma unroll
  for (int r = 0; r < 8; ++r) z[r] = zp[r];
  float* ob = outm + ((size_t)b * NN + n0 + hi * 8) * 128 + cb0 * 16 + lj;
#pragma unroll
  for (int t = 0; t < 4; ++t) {
#pragma unroll
    for (int r = 0; r < 8; ++r) {
      float v = acc[t][r] * z[r];
      v = v > 0.f ? v : (__expf(v) - 1.f);
      ob[(size_t)r * 128 + t * 16] = v;
    }
  }
}

// ---------------- out_p = elu( attn^T @ h_m ) ----------------
__global__ __launch_bounds__(256) void attn_p_kernel(
    const float* __restrict__ sm, const float* __restrict__ sp,
    const float* __restrict__ mxp, const float* __restrict__ izv,
    const float* __restrict__ hmT, float* __restrict__ outp) {
  __shared__ __align__(16) float bufA[DD * LROW];
  __shared__ __align__(16) float bufB[DD * LROW];
  int wave = threadIdx.x >> 5, lane = threadIdx.x & 31;
  int hi = lane >> 4, lj = lane & 15;
  int b = blockIdx.x >> 5;                       // 32 blocks per batch
  int tloc = (blockIdx.x & 31) * 4 + (wave >> 1);
  int cb0 = (wave & 1) * 4;
  int m0 = tloc * 16;
  const float* smb = sm + b * NN;
  const float* zb = izv + b * NN;
  const float* hm = hmT + (size_t)b * 128 * NN;
  float spr = sp[b * MM + m0 + lj];
  float mp = mxp[b];
  unsigned int l0 = (unsigned int)(uintptr_t)bufA;
  unsigned int l1 = (unsigned int)(uintptr_t)bufB;
  if (wave == 0) tdm_load_2d(l0, hm, NN, DD, NN);
  v8f acc[4] = {};
  const int NCH = NN / KC;  // 16
  for (int c = 0; c < NCH; ++c) {
    if (wave == 0) {
      if (c + 1 < NCH) {
        tdm_load_2d(((c + 1) & 1) ? l1 : l0, hm + (c + 1) * KC, NN, DD, NN);
        __builtin_amdgcn_s_wait_tensorcnt(1);
      } else {
        __builtin_amdgcn_s_wait_tensorcnt(0);
      }
    }
    __syncthreads();
    const float* buf = (c & 1) ? bufB : bufA;
    int kbase = c * KC;
    for (int kk = 0; kk < KC; kk += 4) {
      v2f smv = *(const v2f*)(smb + kbase + kk + 2 * hi);
      v2f zv = *(const v2f*)(zb + kbase + kk + 2 * hi);
      float rm0 = leaky(smv.x + mp);
      float rm1 = leaky(smv.y + mp);
      float x0 = leaky(smv.x + spr);
      float x1 = leaky(smv.y + spr);
      v2f a;
      a.x = __expf(x0 - rm0) * zv.x;
      a.y = __expf(x1 - rm1) * zv.y;
#pragma unroll
      for (int t = 0; t < 4; ++t) {
        v2f bf = *(const v2f*)(buf + ((cb0 + t) * 16 + lj) * LROW + kk + 2 * hi);
        acc[t] = __builtin_amdgcn_wmma_f32_16x16x4_f32(false, a, false, bf, (short)0,
                                                       acc[t], false, false);
      }
    }
    __syncthreads();
  }
  float* ob = outp + ((size_t)b * MM + m0 + hi * 8) * 128 + cb0 * 16 + lj;
#pragma unroll
  for (int t = 0; t < 4; ++t) {
#pragma unroll
    for (int r = 0; r < 8; ++r) {
      float v = acc[t][r];
      v = v > 0.f ? v : (__expf(v) - 1.f);
      ob[(size_t)r * 128 + t * 16] = v;
    }
  }
}

extern "C" void kernel_launch(void* const* d_in, const int* in_sizes, int n_in,
                              void* d_out, int out_size, void* d_ws, size_t ws_size,
                              hipStream_t stream) {
  (void)in_sizes; (void)n_in; (void)out_size; (void)ws_size;
  const float* Xm = (const float*)d_in[0];   // (16,1024,128)
  const float* Xp = (const float*)d_in[1];   // (16,2048,128)
  const float* Wm = (const float*)d_in[2];   // (128,128)
  const float* Wp = (const float*)d_in[3];   // (128,128)
  const float* bv = (const float*)d_in[4];   // (256,1)

  float* outm = (float*)d_out;                       // (16,1024,128)
  float* outp = outm + (size_t)BB * NN * DD;         // (16,2048,128)

  float* ws = (float*)d_ws;
  float* hpT = ws;                                   // 16*128*2048
  float* hmT = hpT + (size_t)BB * DD * MM;           // 16*128*1024
  float* smv = hmT + (size_t)BB * DD * NN;           // 16*1024
  float* spv = smv + BB * NN;                        // 16*2048
  float* izv = spv + BB * MM;                        // 16*1024
  float* mxp = izv + BB * NN;                        // 16 (padded to 64)
  float* wbm = mxp + 64;                             // 128
  float* wbp = wbm + 128;                            // 128

  wb_kernel<<<1, 256, 0, stream>>>(Wm, Wp, bv, wbm, wbp);
  s_kernel<<<BB * NN / 8, 256, 0, stream>>>(Xm, wbm, smv);
  s_kernel<<<BB * MM / 8, 256, 0, stream>>>(Xp, wbp, spv);
  maxp_kernel<<<BB, 256, 0, stream>>>(spv, mxp);
  z_kernel<<<BB * NN / 256, 256, 0, stream>>>(smv, spv, mxp, izv);
  proj_kernel<<<BB * NN / 128, 256, 0, stream>>>(Xm, Wm, hmT, NN);
  proj_kernel<<<BB * MM / 128, 256, 0, stream>>>(Xp, Wp, hpT, MM);
  attn_m_kernel<<<BB * 16, 256, 0, stream>>>(smv, spv, mxp, izv, hpT, outm);
  attn_p_kernel<<<BB * 32, 256, 0, stream>>>(smv, spv, mxp, izv, hmT, outp);
}